// PyGHyperGINConv_27831388078179
// MI455X (gfx1250) — compile-verified
//
#include <hip/hip_runtime.h>

typedef __attribute__((ext_vector_type(2))) float v2f;
typedef __attribute__((ext_vector_type(8))) float v8f;

#define NV   50000
#define EV   20000
#define NNZ_ 320000
#define INC  256
#define DD   256

// ---------------------------------------------------------------------------
// GEMM: Xp[NV,DD] = X[NV,INC] @ W[INC,DD]   (FP32 WMMA 16x16x4)
// One wave computes a 16(M) x 64(N) tile: 4 accumulators, A reused 4x.
// A layout (16x4 f32): lane&15 = M row; lane>>4 selects K pair {0,1}/{2,3};
//   a[0]=A[m][kb], a[1]=A[m][kb+1].
// B layout (4x16 f32): lane&15 = N col; b[0]=B[kb][n], b[1]=B[kb+1][n].
// C/D layout (16x16 f32): VGPR v -> row v (lanes 0-15) / row v+8 (lanes 16-31).
// ---------------------------------------------------------------------------
__global__ __launch_bounds__(256) void hgin_gemm_kernel(
    const float* __restrict__ X, const float* __restrict__ W,
    float* __restrict__ Xp) {
  const int wave   = (blockIdx.x * blockDim.x + threadIdx.x) >> 5;
  const int nGroup = DD / 64;                 // 4 N-groups per M-tile row
  const int tileM  = wave / nGroup;
  const int group  = wave % nGroup;
  if (tileM >= NV / 16) return;               // wave-uniform guard (EXEC stays full)

  const int lane = threadIdx.x & 31;
  const int half = lane >> 4;                 // 0: K={0,1}, 1: K={2,3}
  const int mn   = lane & 15;                 // M for A-loads, N for B-loads
  const int kb   = half * 2;
  const int rowA = tileM * 16 + mn;
  const int col0 = group * 64;

  v8f acc0 = {}, acc1 = {}, acc2 = {}, acc3 = {};

  for (int k = 0; k < INC; k += 4) {
    v2f a = *(const v2f*)(X + (size_t)rowA * INC + (k + kb));
    const float* w0 = W + (size_t)(k + kb)     * DD + col0 + mn;
    const float* w1 = W + (size_t)(k + kb + 1) * DD + col0 + mn;
    v2f b0 = { w0[ 0], w1[ 0] };
    v2f b1 = { w0[16], w1[16] };
    v2f b2 = { w0[32], w1[32] };
    v2f b3 = { w0[48], w1[48] };
    acc0 = __builtin_amdgcn_wmma_f32_16x16x4_f32(false, a, false, b0, (short)0, acc0, false, false);
    acc1 = __builtin_amdgcn_wmma_f32_16x16x4_f32(false, a, false, b1, (short)0, acc1, false, false);
    acc2 = __builtin_amdgcn_wmma_f32_16x16x4_f32(false, a, false, b2, (short)0, acc2, false, false);
    ac3:
    acc3 = __builtin_amdgcn_wmma_f32_16x16x4_f32(false, a, false, b3, (short)0, acc3, false, false);
  }

  float* outBase = Xp + (size_t)(tileM * 16) * DD + col0 + mn;
  #pragma unroll
  for (int v = 0; v < 8; ++v) {
    const int row = v + half * 8;
    float* o = outBase + (size_t)row * DD;
    o[ 0] = acc0[v];
    o[16] = acc1[v];
    o[32] = acc2[v];
    o[48] = acc3[v];
  }
}

// ---------------------------------------------------------------------------
// Zero a float buffer (float4 vectorized; n4 = count/4).
// ---------------------------------------------------------------------------
__global__ __launch_bounds__(256) void hgin_zero_kernel(float4* __restrict__ p, int n4) {
  int i = blockIdx.x * blockDim.x + threadIdx.x;
  if (i < n4) p[i] = make_float4(0.f, 0.f, 0.f, 0.f);
}

// ---------------------------------------------------------------------------
// out = (1 + eps) * Xp, float4 vectorized.
// ---------------------------------------------------------------------------
__global__ __launch_bounds__(256) void hgin_scale_kernel(
    const float* __restrict__ Xp, const float* __restrict__ eps,
    float* __restrict__ out, int n4) {
  int i = blockIdx.x * blockDim.x + threadIdx.x;
  if (i >= n4) return;
  const float s = 1.0f + eps[0];
  float4 v = ((const float4*)Xp)[i];
  v.x *= s; v.y *= s; v.z *= s; v.w *= s;
  ((float4*)out)[i] = v;
}

// ---------------------------------------------------------------------------
// Generic incidence scatter: dst[di[i], :] += src[si[i], :]
// One wave per incidence. Lane owns 8 contiguous channels [lane*8, lane*8+8):
// 2 x global_load_b128 per lane (512 B per wave per load instruction), then
// 8 hardware f32 atomic adds (L2-resident destinations).
// ---------------------------------------------------------------------------
__device__ __forceinline__ void scatter_row_add(const float* __restrict__ src,
                                                float* __restrict__ dst,
                                                int lane) {
  const float4* s4 = (const float4*)(src + lane * 8);
  float4 a = s4[0];
  float4 b = s4[1];
  float* d = dst + lane * 8;
  unsafeAtomicAdd(d + 0, a.x);
  unsafeAtomicAdd(d + 1, a.y);
  unsafeAtomicAdd(d + 2, a.z);
  unsafeAtomicAdd(d + 3, a.w);
  unsafeAtomicAdd(d + 4, b.x);
  unsafeAtomicAdd(d + 5, b.y);
  unsafeAtomicAdd(d + 6, b.z);
  unsafeAtomicAdd(d + 7, b.w);
}

// Scatter vertex->edge: Xe[edges[i], :] += Xp[vertex[i], :]
__global__ __launch_bounds__(256) void hgin_scatter_v2e_kernel(
    const float* __restrict__ Xp, const int* __restrict__ vertex,
    const int* __restrict__ edges, float* __restrict__ Xe) {
  const int inc = (blockIdx.x * blockDim.x + threadIdx.x) >> 5;
  if (inc >= NNZ_) return;
  const int lane = threadIdx.x & 31;
  scatter_row_add(Xp + (size_t)vertex[inc] * DD,
                  Xe + (size_t)edges[inc]  * DD, lane);
}

// Scatter edge->vertex: out[vertex[i], :] += Xe[edges[i], :]
__global__ __launch_bounds__(256) void hgin_scatter_e2v_kernel(
    const float* __restrict__ Xe, const int* __restrict__ vertex,
    const int* __restrict__ edges, float* __restrict__ out) {
  const int inc = (blockIdx.x * blockDim.x + threadIdx.x) >> 5;
  if (inc >= NNZ_) return;
  const int lane = threadIdx.x & 31;
  scatter_row_add(Xe  + (size_t)edges[inc]  * DD,
                  out + (size_t)vertex[inc] * DD, lane);
}

// ---------------------------------------------------------------------------
extern "C" void kernel_launch(void* const* d_in, const int* in_sizes, int n_in,
                              void* d_out, int out_size, void* d_ws, size_t ws_size,
                              hipStream_t stream) {
  const float* X      = (const float*)d_in[0];
  const int*   vertex = (const int*)  d_in[1];
  const int*   edges  = (const int*)  d_in[2];
  const float* W      = (const float*)d_in[3];
  const float* eps    = (const float*)d_in[4];
  float*       out    = (float*)d_out;

  float* Xp = (float*)d_ws;                       // NV*DD floats = 51.2 MB
  float* Xe = Xp + (size_t)NV * DD;               // EV*DD floats = 20.5 MB

  // 1) Zero edge accumulator.
  {
    const int n4 = EV * DD / 4;
    hgin_zero_kernel<<<(n4 + 255) / 256, 256, 0, stream>>>((float4*)Xe, n4);
  }
  // 2) Xp = X @ W  (FP32 WMMA). 12500 wave-tiles, 8 waves per 256-thread block.
  {
    const int waves = (NV / 16) * (DD / 64);
    hgin_gemm_kernel<<<(waves + 7) / 8, 256, 0, stream>>>(X, W, Xp);
  }
  // 3) out = (1+eps) * Xp.
  {
    const int n4 = NV * DD / 4;
    hgin_scale_kernel<<<(n4 + 255) / 256, 256, 0, stream>>>(Xp, eps, out, n4);
  }
  // 4) Xe += scatter(Xp by vertex -> edges).
  hgin_scatter_v2e_kernel<<<(NNZ_ + 7) / 8, 256, 0, stream>>>(Xp, vertex, edges, Xe);
  // 5) out += scatter(Xe by edges -> vertex).
  hgin_scatter_e2v_kernel<<<(NNZ_ + 7) / 8, 256, 0, stream>>>(Xe, vertex, edges, out);
}